// Attention_67826123538785
// MI455X (gfx1250) — compile-verified
//
#include <hip/hip_runtime.h>
#include <hip/hip_bf16.h>

typedef float v2f __attribute__((ext_vector_type(2)));
typedef float v8f __attribute__((ext_vector_type(8)));

#define NKEYS 400   // ne*ne
#define LDK   68    // padded LDS row stride (floats) for K/V chunk & Q staging
#define LDP   20    // padded LDS row stride (floats) for P scratch

__device__ __forceinline__ v8f wmma4(v2f a, v2f b, v8f c) {
  // D = A(16x4) * B(4x16) + C, fp32 WMMA (wave32)
  return __builtin_amdgcn_wmma_f32_16x16x4_f32(false, a, false, b, (short)0, c,
                                               false, false);
}

__global__ __launch_bounds__(256) void attn_fused_kernel(
    const float* __restrict__ R,   const float* __restrict__ Rmas,
    const float* __restrict__ WQ,  const float* __restrict__ WQb,
    const float* __restrict__ WK,  const float* __restrict__ WKb,
    const float* __restrict__ WV,  const float* __restrict__ WVb,
    float* __restrict__ Out)
{
  // union: [Q staging: 8 waves x 16 x LDK]  OR  [Ks: 64 x LDK | Vs: 64 x LDK]
  __shared__ float sKV[2 * 64 * LDK];          // 34,816 B
  __shared__ float sP[8][16 * LDP];            // 10,240 B

  const int lane = threadIdx.x & 31;
  const int wave = threadIdx.x >> 5;
  const int n    = lane & 15;                  // column / row-within-16
  const int hi   = lane >> 4;                  // half-wave select

  const int qcg = blockIdx.x;                  // query chunk 0..3
  const int h   = blockIdx.y;                  // head 0..11
  const int b   = blockIdx.z;                  // batch 0..31

  const float* Rbh  = R + (size_t)b * NKEYS * 768 + h * 64;
  const float* mrow = Rmas + (size_t)b * NKEYS;

  int qtile = qcg * 8 + wave;                  // 16-row query tile
  const bool writer = (qtile < 25);
  if (qtile > 24) qtile = 24;                  // clamp: uniform control flow
  const int q16 = qtile * 16;

  // ---------------- Q = R_q @ WQ^T + b (4 col-tiles of 16) ----------------
  v8f qacc[4];
  #pragma unroll
  for (int t = 0; t < 4; ++t) {
    v8f acc = {};
    const float* arow = Rbh + (size_t)(q16 + n) * 768 + 2 * hi;   // A row = n
    const float* brow = WQ + (t * 16 + n) * 64 + 2 * hi;          // B[c][o]=WQ[o][c]
    #pragma unroll
    for (int s = 0; s < 16; ++s) {
      v2f a  = *(const v2f*)(arow + 4 * s);
      v2f bb = *(const v2f*)(brow + 4 * s);
      acc = wmma4(a, bb, acc);
    }
    float bv = WQb[t * 16 + n];
    #pragma unroll
    for (int v = 0; v < 8; ++v) acc[v] += bv;
    qacc[t] = acc;
  }
  // C-layout -> A-layout via per-wave LDS staging (region later reused as K/V)
  float* qs = sKV + wave * (16 * LDK);
  #pragma unroll
  for (int t = 0; t < 4; ++t)
    #pragma unroll
    for (int v = 0; v < 8; ++v)
      qs[(v + 8 * hi) * LDK + t * 16 + n] = qacc[t][v];
  __syncthreads();
  v2f qa[16];                                   // Q tile as A fragments
  #pragma unroll
  for (int s = 0; s < 16; ++s)
    qa[s] = *(const v2f*)(qs + n * LDK + 4 * s + 2 * hi);
  __syncthreads();                              // staging area free for K/V

  float mq[8];
  #pragma unroll
  for (int v = 0; v < 8; ++v) mq[v] = mrow[q16 + v + 8 * hi];

  v8f   oacc[4] = {};                           // O accumulator 16x64
  float m_run[8], l_run[8];
  #pragma unroll
  for (int v = 0; v < 8; ++v) { m_run[v] = -1e30f; l_run[v] = 0.0f; }

  float* Ks = sKV;
  float* Vs = sKV + 64 * LDK;
  float* Pw = &sP[wave][0];

  // ---------------- stream keys in chunks of 64 ----------------
  for (int ck = 0; ck < 7; ++ck) {
    const int kt0   = ck * 4;
    const int nkt   = (25 - kt0) < 4 ? (25 - kt0) : 4;   // key tiles this chunk
    const int kbase = kt0 * 16;

    // K/V projection into LDS: waves 0-3 -> K (col tiles 0-3), 4-7 -> V
    {
      const int mat = wave >> 2;
      const int nt  = wave & 3;
      const float* W  = mat ? WV  : WK;
      const float* Bb = mat ? WVb : WKb;
      float* dst      = mat ? Vs  : Ks;
      for (int mt = 0; mt < nkt; ++mt) {
        v8f acc = {};
        const float* arow = Rbh + (size_t)(kbase + mt * 16 + n) * 768 + 2 * hi;
        const float* brow = W + (nt * 16 + n) * 64 + 2 * hi;
        #pragma unroll
        for (int s = 0; s < 16; ++s) {
          v2f a  = *(const v2f*)(arow + 4 * s);
          v2f bb = *(const v2f*)(brow + 4 * s);
          acc = wmma4(a, bb, acc);
        }
        float bv = Bb[nt * 16 + n];
        #pragma unroll
        for (int v = 0; v < 8; ++v)
          dst[(mt * 16 + v + 8 * hi) * LDK + nt * 16 + n] = acc[v] + bv;
      }
    }
    __syncthreads();

    // attention against this chunk (all waves: uniform barriers)
    for (int ktl = 0; ktl < nkt; ++ktl) {
      // S = Q @ K_tile^T   (B[c][key]=Ks[key][c])
      v8f s = {};
      const float* krow = Ks + (ktl * 16 + n) * LDK + 2 * hi;
      #pragma unroll
      for (int st = 0; st < 16; ++st) {
        v2f bb = *(const v2f*)(krow + 4 * st);
        s = wmma4(qa[st], bb, s);
      }
      // mask BEFORE scaling (faithful), then 1/sqrt(64)
      const float mk = mrow[kbase + ktl * 16 + n];
      #pragma unroll
      for (int v = 0; v < 8; ++v)
        s[v] = (s[v] - (1.0f - mq[v] * mk) * 100000.0f) * 0.125f;

      // online softmax (row = v + 8*hi, reduce across 16 lanes)
      #pragma unroll
      for (int v = 0; v < 8; ++v) {
        float x = s[v];
        x = fmaxf(x, __shfl_xor(x, 1, 16));
        x = fmaxf(x, __shfl_xor(x, 2, 16));
        x = fmaxf(x, __shfl_xor(x, 4, 16));
        x = fmaxf(x, __shfl_xor(x, 8, 16));
        const float nm   = fmaxf(m_run[v], x);
        const float corr = __expf(m_run[v] - nm);
        const float p    = __expf(s[v] - nm);
        float ts = p;
        ts += __shfl_xor(ts, 1, 16);
        ts += __shfl_xor(ts, 2, 16);
        ts += __shfl_xor(ts, 4, 16);
        ts += __shfl_xor(ts, 8, 16);
        l_run[v] = l_run[v] * corr + ts;
        m_run[v] = nm;
        #pragma unroll
        for (int t = 0; t < 4; ++t) oacc[t][v] *= corr;
        Pw[(v + 8 * hi) * LDP + n] = p;          // P scratch (C-layout)
      }
      __syncthreads();

      // O += P @ V_tile  (K=16, 4 steps of 4)
      #pragma unroll
      for (int st = 0; st < 4; ++st) {
        v2f a = *(const v2f*)(Pw + n * LDP + 4 * st + 2 * hi);
        #pragma unroll
        for (int t = 0; t < 4; ++t) {
          const float* vp = Vs + (ktl * 16 + 4 * st + 2 * hi) * LDK + t * 16 + n;
          v2f bb; bb.x = vp[0]; bb.y = vp[LDK];
          oacc[t] = wmma4(a, bb, oacc[t]);
        }
      }
      __syncthreads();
    }
    __syncthreads();   // all reads of Ks/Vs done before next chunk overwrites
  }

  // ---------------- epilogue: /l, * query mask, store ----------------
  if (writer) {
    #pragma unroll
    for (int v = 0; v < 8; ++v) {
      const float scale = mq[v] / l_run[v];
      const size_t row = (size_t)(b * NKEYS + q16 + v + 8 * hi) * 768 + h * 64;
      #pragma unroll
      for (int t = 0; t < 4; ++t)
        Out[row + t * 16 + n] = oacc[t][v] * scale;
    }
  }
}

extern "C" void kernel_launch(void* const* d_in, const int* in_sizes, int n_in,
                              void* d_out, int out_size, void* d_ws, size_t ws_size,
                              hipStream_t stream) {
  (void)in_sizes; (void)n_in; (void)out_size; (void)d_ws; (void)ws_size;
  const float* R    = (const float*)d_in[0];
  const float* Rmas = (const float*)d_in[1];
  const float* WQ   = (const float*)d_in[2];
  const float* WQb  = (const float*)d_in[3];
  const float* WK   = (const float*)d_in[4];
  const float* WKb  = (const float*)d_in[5];
  const float* WV   = (const float*)d_in[6];
  const float* WVb  = (const float*)d_in[7];
  float* out = (float*)d_out;

  dim3 grid(4, 12, 32);   // (query-chunk, head, batch)
  dim3 block(256);        // 8 waves (wave32)
  hipLaunchKernelGGL(attn_fused_kernel, grid, block, 0, stream,
                     R, Rmas, WQ, WQb, WK, WKb, WV, WVb, out);
}